// CustomLoss_46024869544057
// MI455X (gfx1250) — compile-verified
//
#include <hip/hip_runtime.h>

typedef __attribute__((ext_vector_type(2))) float v2f;
typedef __attribute__((ext_vector_type(8))) float v8f;

#define THREADS 256
#define BLOCKS  1024

// Wave32 all-reduce via V_WMMA_F32_16X16X4_F32.
// A: lane's value in slot0, 0 in slot1 (64 slots total across wave).
// B: all ones  => D[m,n] = rowsum(A[m,:]) for every column n.
// Summing a full D column gives the sum of ALL A slots. Per the documented
// C/D layout, lanes 0-15 hold rows 0-7 (8 VGPRs) and lanes 16-31 hold rows
// 8-15, so per-lane sum of the 8 D regs + xor-16 shuffle = wave total,
// independent of A's exact striping (B is all ones).
__device__ __forceinline__ float wave_allreduce_f32(float x) {
    v2f a; a[0] = x;    a[1] = 0.0f;
    v2f b; b[0] = 1.0f; b[1] = 1.0f;
    v8f c = {};
    v8f d = __builtin_amdgcn_wmma_f32_16x16x4_f32(
        /*neg_a=*/false, a, /*neg_b=*/false, b,
        /*c_mod=*/(short)0, c, /*reuse_a=*/false, /*reuse_b=*/false);
    float s = ((d[0] + d[1]) + (d[2] + d[3])) + ((d[4] + d[5]) + (d[6] + d[7]));
    s += __shfl_xor(s, 16, 32);   // combine rows 0-7 with rows 8-15
    return s;                     // wave total in every lane
}

// Reduce (loss, count) across the block; thread 0 gets the totals.
__device__ __forceinline__ bool block_reduce2(float l, float cnt,
                                              float* totl, float* totc) {
    __shared__ float lsh[THREADS / 32];
    __shared__ float csh[THREADS / 32];
    float lw = wave_allreduce_f32(l);
    float cw = wave_allreduce_f32(cnt);
    const int lane = threadIdx.x & 31;
    const int w    = threadIdx.x >> 5;
    if (lane == 0) { lsh[w] = lw; csh[w] = cw; }
    __syncthreads();
    if (threadIdx.x == 0) {
        float L = 0.0f, C = 0.0f;
        #pragma unroll
        for (int i = 0; i < THREADS / 32; ++i) { L += lsh[i]; C += csh[i]; }
        *totl = L; *totc = C;
        return true;
    }
    return false;
}

__global__ void __launch_bounds__(THREADS)
bce_stage1(const float* __restrict__ in, const int* __restrict__ tg,
           float* __restrict__ ws, int nvec, int n) {
    const float4* __restrict__ in4 = (const float4*)in;
    const int4*   __restrict__ tg4 = (const int4*)tg;
    const int gid    = blockIdx.x * THREADS + threadIdx.x;
    const int stride = gridDim.x * THREADS;

    float lsum = 0.0f;   // sum of log(q), q = t ? p : 1-p  (negated at the end)
    float csum = 0.0f;   // count of (p > 0.5 && t == 0), exactly integral

    for (int i = gid; i < nvec; i += stride) {
        float4 p = in4[i];
        int4   t = tg4[i];
        float q0 = t.x ? p.x : 1.0f - p.x;
        float q1 = t.y ? p.y : 1.0f - p.y;
        float q2 = t.z ? p.z : 1.0f - p.z;
        float q3 = t.w ? p.w : 1.0f - p.w;
        lsum += (__logf(q0) + __logf(q1)) + (__logf(q2) + __logf(q3));
        csum += ((p.x > 0.5f) && (t.x == 0)) ? 1.0f : 0.0f;
        csum += ((p.y > 0.5f) && (t.y == 0)) ? 1.0f : 0.0f;
        csum += ((p.z > 0.5f) && (t.z == 0)) ? 1.0f : 0.0f;
        csum += ((p.w > 0.5f) && (t.w == 0)) ? 1.0f : 0.0f;
    }

    // Scalar tail (n % 4), handled by one thread; N=2^24 so normally empty.
    if (gid == 0) {
        for (int i = nvec * 4; i < n; ++i) {
            float p = in[i]; int t = tg[i];
            lsum += __logf(t ? p : 1.0f - p);
            csum += ((p > 0.5f) && (t == 0)) ? 1.0f : 0.0f;
        }
    }

    float L, C;
    if (block_reduce2(lsum, csum, &L, &C)) {
        ws[blockIdx.x]          = L;
        ws[BLOCKS + blockIdx.x] = C;
    }
}

__global__ void __launch_bounds__(THREADS)
bce_stage2(const float* __restrict__ ws, float* __restrict__ out,
           float inv_n, int nblocks) {
    float lsum = 0.0f, csum = 0.0f;
    for (int i = threadIdx.x; i < nblocks; i += THREADS) {
        lsum += ws[i];
        csum += ws[nblocks + i];
    }
    float L, C;
    if (block_reduce2(lsum, csum, &L, &C)) {
        float mean       = -L * inv_n;            // BCE mean
        float multiplier = 1.0f + 0.1f * C;       // matches reference f32 math
        out[0] = mean * multiplier;
    }
}

extern "C" void kernel_launch(void* const* d_in, const int* in_sizes, int n_in,
                              void* d_out, int out_size, void* d_ws, size_t ws_size,
                              hipStream_t stream) {
    const float* in = (const float*)d_in[0];
    const int*   tg = (const int*)d_in[1];
    float*       out = (float*)d_out;
    float*       ws  = (float*)d_ws;   // needs 2*BLOCKS floats = 8 KB

    const int n    = in_sizes[0];
    const int nvec = n / 4;

    bce_stage1<<<BLOCKS, THREADS, 0, stream>>>(in, tg, ws, nvec, n);
    bce_stage2<<<1, THREADS, 0, stream>>>(ws, out, 1.0f / (float)n, BLOCKS);
}